// FISM_55284819034149
// MI455X (gfx1250) — compile-verified
//
#include <hip/hip_runtime.h>
#include <hip/hip_bf16.h>

// FISM scoring for MI455X (gfx1250, wave32).
// Kernel 1: masked sum-pool of past-item embeddings -> profile[B,64] (in d_ws)
// Kernel 2: per-candidate dot via chained V_WMMA_F32_16X16X4_F32 + bias.

#define B_ROWS   4096
#define N_DOCS   200
#define HIST     200
#define NFAC     64

typedef __attribute__((ext_vector_type(2))) float v2f;
typedef __attribute__((ext_vector_type(8))) float v8f;

// ---------------------------------------------------------------------------
// Kernel 1: pooled[b] = (sum_h mask(h) * past_item_table[past[b,h]]) * len^-0.5
// One block (256 threads = 8 waves) per batch row. Thread t: feature f = t&63,
// HIST slice part = t>>6. Wave-uniform h loop -> scalar index loads + scalar
// branch around the (coalesced, 128B/wave) row load.
// ---------------------------------------------------------------------------
__global__ void fism_profile_kernel(const int* __restrict__ past_item_lst,
                                    const int* __restrict__ len_past_items,
                                    const float* __restrict__ past_item_table,
                                    float* __restrict__ profile) {
    const int b    = blockIdx.x;
    const int t    = threadIdx.x;
    const int f    = t & (NFAC - 1);
    const int part = t >> 6;                       // 0..3

    const int* hist = past_item_lst + (size_t)b * HIST;
    float acc = 0.0f;
    const int h0 = part * (HIST / 4);
    #pragma unroll 5
    for (int h = h0; h < h0 + HIST / 4; ++h) {
        const int idx = hist[h];                   // wave-uniform -> scalar branch
        if (idx > 0)
            acc += past_item_table[(size_t)idx * NFAC + f];
    }

    __shared__ float red[4][NFAC];
    red[part][f] = acc;
    __syncthreads();

    if (t < NFAC) {
        const float s = red[0][t] + red[1][t] + red[2][t] + red[3][t];
        const float coeff = rsqrtf((float)len_past_items[b]);  // len^-ALPHA, ALPHA=0.5
        profile[(size_t)b * NFAC + t] = s * coeff;
    }
}

// ---------------------------------------------------------------------------
// Kernel 2: scores[b,d] = dot(profile[b], mask(d)*item_table[item_lst[b,d]])
//                         + item_bias_table[item_lst[b,d]]
// One block per batch row; each wave owns 16-doc tiles. F=64 -> 16 chained
// V_WMMA_F32_16X16X4_F32.  A (16x4 f32): lane L (row=L&15, hi=L>>4) supplies
// float2 of doc row at column 4*kc + 2*hi.  B (4x16): profile chunk broadcast
// across all 16 columns -> every column of D is the same dot product.
//
// Masking is applied as DATA (multiply by 0/1), never as EXEC control flow:
// all 16 b64 gathers of a row are issued unconditionally and in a single
// batch so the s_wait_loadcnt before each WMMA is a partial count, keeping
// the full gather in flight (this kernel is HBM-gather bound, ~420 MB total).
//
// D layout: VGPR r, lanes 0-15 -> M=r; lanes 16-31 -> M=r+8.  Lanes 0 and 16
// write 8 scores each (bias gathered cross-lane via __shfl).
// ---------------------------------------------------------------------------
__global__ void fism_score_kernel(const int* __restrict__ item_lst,
                                  const float* __restrict__ item_table,
                                  const float* __restrict__ item_bias_table,
                                  const float* __restrict__ profile,
                                  float* __restrict__ out) {
    const int b = blockIdx.x;
    const int t = threadIdx.x;

    __shared__ float prof[NFAC];
    if (t < NFAC) prof[t] = profile[(size_t)b * NFAC + t];
    __syncthreads();

    const int wave = t >> 5;        // 0..7
    const int lane = t & 31;
    const int row  = lane & 15;     // doc within tile (M)
    const int hi   = lane >> 4;     // K half-select

    // Hoist the (tile-invariant) B operand chunks into registers.
    v2f bv[16];
    #pragma unroll
    for (int kc = 0; kc < 16; ++kc) {
        bv[kc].x = prof[kc * 4 + hi * 2 + 0];
        bv[kc].y = prof[kc * 4 + hi * 2 + 1];
    }

    for (int tile = wave; tile * 16 < N_DOCS; tile += 8) {   // tiles 0..12
        const int  doc   = tile * 16 + row;
        const bool inRng = doc < N_DOCS;
        const int  idx   = inRng ? item_lst[(size_t)b * N_DOCS + doc] : 0;
        const bool valid = inRng && (idx > 0);     // reference masks emb only
        const float m    = valid ? 1.0f : 0.0f;    // data mask, no EXEC branches
        const float bias = inRng ? item_bias_table[idx] : 0.0f;

        // idx >= 0 always, so an unmasked gather of row `idx` is safe.
        const float* arow = item_table + (size_t)idx * NFAC + hi * 2;

        // Issue ALL 16 b64 gathers of this lane's half-row up front.
        v2f a[16];
        #pragma unroll
        for (int kc = 0; kc < 16; ++kc) {
            a[kc].x = arow[kc * 4 + 0];            // contiguous float2 -> b64 load
            a[kc].y = arow[kc * 4 + 1];
        }

        v8f c = {0.f, 0.f, 0.f, 0.f, 0.f, 0.f, 0.f, 0.f};
        #pragma unroll
        for (int kc = 0; kc < 16; ++kc) {
            v2f am;
            am.x = a[kc].x * m;
            am.y = a[kc].y * m;
            c = __builtin_amdgcn_wmma_f32_16x16x4_f32(
                    /*neg_a=*/false, am, /*neg_b=*/false, bv[kc],
                    /*c_mod=*/(short)0, c, /*reuse_a=*/false, /*reuse_b=*/false);
        }

        // Lanes 0 (M=0..7) and 16 (M=8..15) write their 8 accumulators.
        #pragma unroll
        for (int r = 0; r < 8; ++r) {
            const int   srcRow = hi * 8 + r;             // doc row whose bias we need
            const float br     = __shfl(bias, srcRow, 32);
            if (row == 0) {
                const int d = tile * 16 + srcRow;
                if (d < N_DOCS)
                    out[(size_t)b * N_DOCS + d] = c[r] + br;
            }
        }
    }
}

// ---------------------------------------------------------------------------
extern "C" void kernel_launch(void* const* d_in, const int* in_sizes, int n_in,
                              void* d_out, int out_size, void* d_ws, size_t ws_size,
                              hipStream_t stream) {
    const int*   item_lst        = (const int*)  d_in[0];
    const int*   past_item_lst   = (const int*)  d_in[1];
    const int*   len_past_items  = (const int*)  d_in[2];
    const float* item_table      = (const float*)d_in[3];
    const float* past_item_table = (const float*)d_in[4];
    const float* item_bias_table = (const float*)d_in[5];
    float*       out             = (float*)d_out;
    float*       profile         = (float*)d_ws;          // B_ROWS*64 floats = 1 MB

    fism_profile_kernel<<<B_ROWS, 256, 0, stream>>>(past_item_lst, len_past_items,
                                                    past_item_table, profile);
    fism_score_kernel<<<B_ROWS, 256, 0, stream>>>(item_lst, item_table,
                                                  item_bias_table, profile, out);
}